// _ConvBlock5_5_43018392436854
// MI455X (gfx1250) — compile-verified
//
#include <hip/hip_runtime.h>
#include <stdint.h>

// Graph unpooling: out[dst] += edge_attr * x[src], C=128 f32 channels.
// Memory-bound scatter-add:
//   - one wave32 per edge: lane i handles channels [4i, 4i+4) via float4
//   - edge indices loaded at wave-uniform addresses -> scalar s_load path (KMcnt)
//   - 4x native global_atomic_add_f32 (no-return, agent scope) per lane,
//     resolved in L2 (out = 102MB, x = 26MB, both fit in the 192MB L2)
//   - 2x unrolled grid-stride: two independent gather->atomic chains per
//     iteration for latency hiding; global_prefetch_b8 leads by 2*stride
//   - zero-fill pass first (harness poisons d_out with 0xAA)

#define CHN 128

__device__ __forceinline__ void atomic_add_row(float* o, float4 v, float w) {
    (void)__hip_atomic_fetch_add(o + 0, v.x * w, __ATOMIC_RELAXED, __HIP_MEMORY_SCOPE_AGENT);
    (void)__hip_atomic_fetch_add(o + 1, v.y * w, __ATOMIC_RELAXED, __HIP_MEMORY_SCOPE_AGENT);
    (void)__hip_atomic_fetch_add(o + 2, v.z * w, __ATOMIC_RELAXED, __HIP_MEMORY_SCOPE_AGENT);
    (void)__hip_atomic_fetch_add(o + 3, v.w * w, __ATOMIC_RELAXED, __HIP_MEMORY_SCOPE_AGENT);
}

__global__ __launch_bounds__(256) void _unpool_scatter_kernel(
        const float* __restrict__ x,        // [N_coarse, 128]
        const int*   __restrict__ src_idx,  // [E]
        const int*   __restrict__ dst_idx,  // [E]
        const float* __restrict__ edge_w,   // [E]
        float*       __restrict__ out,      // [N_fine, 128]
        int E) {
    const int lane = threadIdx.x & 31;
    const int waves_per_block = blockDim.x >> 5;
    // Force the edge id into an SGPR so per-edge index/weight loads become
    // scalar loads (constant cache / KMcnt) instead of 32 redundant VMEM lanes.
    int wave = __builtin_amdgcn_readfirstlane(
        (int)(blockIdx.x * waves_per_block + (threadIdx.x >> 5)));
    const int stride  = gridDim.x * waves_per_block;
    const int stride2 = stride * 2;

    for (int e = wave; e < E; e += stride2) {
        const int e1 = e + stride;

        // Prefetch the gathered rows two/three strides ahead
        // (gfx1250 global_prefetch_b8) to cover L2 gather latency.
        const int ep0 = e + stride2;
        if (ep0 < E) {
            __builtin_prefetch((const void*)(x + (size_t)src_idx[ep0] * CHN), 0, 3);
        }
        const int ep1 = ep0 + stride;
        if (ep1 < E) {
            __builtin_prefetch((const void*)(x + (size_t)src_idx[ep1] * CHN), 0, 3);
        }

        const int   s0 = src_idx[e];     // wave-uniform -> s_load
        const int   d0 = dst_idx[e];
        const float w0 = edge_w[e];

        if (e1 < E) {                    // wave-uniform branch (s_cbranch)
            const int   s1 = src_idx[e1];
            const int   d1 = dst_idx[e1];
            const float w1 = edge_w[e1];

            // Two independent coalesced 16B-per-lane gathers (global_load_b128);
            // each wave covers a full 512B row.
            const float4 v0 = ((const float4*)(x + (size_t)s0 * CHN))[lane];
            const float4 v1 = ((const float4*)(x + (size_t)s1 * CHN))[lane];

            atomic_add_row(out + (size_t)d0 * CHN + lane * 4, v0, w0);
            atomic_add_row(out + (size_t)d1 * CHN + lane * 4, v1, w1);
        } else {
            const float4 v0 = ((const float4*)(x + (size_t)s0 * CHN))[lane];
            atomic_add_row(out + (size_t)d0 * CHN + lane * 4, v0, w0);
        }
    }
}

__global__ __launch_bounds__(256) void _unpool_zero_kernel(float4* __restrict__ out4,
                                                           int n4) {
    int i = blockIdx.x * blockDim.x + threadIdx.x;
    if (i < n4) {
        out4[i] = make_float4(0.f, 0.f, 0.f, 0.f);
    }
}

extern "C" void kernel_launch(void* const* d_in, const int* in_sizes, int n_in,
                              void* d_out, int out_size, void* d_ws, size_t ws_size,
                              hipStream_t stream) {
    const float* x        = (const float*)d_in[0];
    const int*   src_idx  = (const int*)d_in[1];
    const int*   dst_idx  = (const int*)d_in[2];
    const float* edge_w   = (const float*)d_in[3];
    float*       out      = (float*)d_out;

    const int E  = in_sizes[1];          // number of edges
    const int n4 = out_size / 4;         // out_size = n_fine * 128, divisible by 4

    // Pass 1: zero the output (float4 stores).
    {
        int threads = 256;
        int blocks  = (n4 + threads - 1) / threads;
        _unpool_zero_kernel<<<blocks, threads, 0, stream>>>((float4*)out, n4);
    }

    // Pass 2: scatter-add. 4096 blocks x 8 waves = 32768 waves, 2x-unrolled
    // grid-stride over E edges (~12 iterations/wave) with prefetch lead.
    {
        int threads = 256;
        int waves_per_block = threads / 32;
        int blocks = 4096;
        int max_blocks = (E + waves_per_block - 1) / waves_per_block;
        if (blocks > max_blocks) blocks = max_blocks;
        _unpool_scatter_kernel<<<blocks, threads, 0, stream>>>(
            x, src_idx, dst_idx, edge_w, out, E);
    }
}